// EmbeddingShard_2826088480925
// MI455X (gfx1250) — compile-verified
//
#include <hip/hip_runtime.h>
#include <stdint.h>

typedef float v4f __attribute__((ext_vector_type(4)));

#define NSHARDS 8
#define VSH     6300
#define DMODEL  4096
#define SEQLEN  2048
#define NBATCH  4

// ---------------------------------------------------------------------------
// Kernel 1: precompute edge[s][d] = sum_{j<s} W[j,VSH-1,d] + sum_{j>s} W[j,0,d]
//           + NSHARDS * bias[d]          (8 x 4096 floats, stored in d_ws)
// ---------------------------------------------------------------------------
__global__ __launch_bounds__(128) void edge_precompute(
    const float* __restrict__ W, const float* __restrict__ bias,
    float* __restrict__ edge) {
  const int d = blockIdx.x * 128 + threadIdx.x;   // 0..DMODEL-1
  float firstRow[NSHARDS], lastRow[NSHARDS];
#pragma unroll
  for (int j = 0; j < NSHARDS; ++j) {
    const size_t base = (size_t)j * VSH * DMODEL;
    firstRow[j] = W[base + (size_t)d];
    lastRow[j]  = W[base + (size_t)(VSH - 1) * DMODEL + (size_t)d];
  }
  const float b8 = (float)NSHARDS * bias[d];
#pragma unroll
  for (int s = 0; s < NSHARDS; ++s) {
    float acc = b8;
#pragma unroll
    for (int j = 0; j < NSHARDS; ++j) {
      if (j < s) acc += lastRow[j];
      if (j > s) acc += firstRow[j];
    }
    edge[s * DMODEL + d] = acc;
  }
}

// ---------------------------------------------------------------------------
// Kernel 2: one block per token.
//   out[bt,:] = W[s*, v - s*·VSH, :]  (async-DMA'd into LDS)
//             + edge[s*]  (L2-resident) + pos_emb[t,:]
// Async row->LDS overlaps with the pos/edge global loads; NT stores keep L2
// free for pos_emb reuse across the batch.
// ---------------------------------------------------------------------------
__global__ __launch_bounds__(256) void gather_add(
    const int* __restrict__ x, const float* __restrict__ W,
    const float* __restrict__ pos, const float* __restrict__ edge,
    float* __restrict__ out) {
  __shared__ v4f smem[DMODEL / 4];                // 16 KB row staging
  const int bt  = (int)blockIdx.x;                // 0..B*SEQ-1
  const int t   = bt & (SEQLEN - 1);
  const int tid = (int)threadIdx.x;

  const int v     = x[bt];                        // uniform per block
  const int s     = v / VSH;                      // home shard, exact in [0,7]
  const int local = v - s * VSH;

  const float* rowBase = W + ((size_t)s * VSH + (size_t)local) * (size_t)DMODEL;
  const v4f* posRow  = (const v4f*)(pos  + (size_t)t * DMODEL);
  const v4f* edgeRow = (const v4f*)(edge + (size_t)s * DMODEL);
  float* outRow = out + (size_t)bt * DMODEL;

#if defined(__HIP_DEVICE_COMPILE__) && defined(__gfx1250__)
  // LDS byte offset of the staging buffer (flat addr low 32 bits == LDS addr)
  const unsigned ldsBase = (unsigned)(uintptr_t)(&smem[0]);
  // Issue 4 async b128 loads per lane: whole 16 KB row -> LDS (ASYNCcnt)
#pragma unroll
  for (int i = 0; i < 4; ++i) {
    const unsigned goff = (unsigned)((tid + i * 256) * 16);  // byte offset in row
    const unsigned loff = ldsBase + goff;
    asm volatile("global_load_async_to_lds_b128 %0, %1, %2"
                 :: "v"(loff), "v"(goff), "s"(rowBase)
                 : "memory");
  }
  // Overlap: pull pos_emb + edge while the row DMA is in flight
  v4f p[4], e[4];
#pragma unroll
  for (int i = 0; i < 4; ++i) {
    const int slot = tid + i * 256;
    p[i] = posRow[slot];
    e[i] = edgeRow[slot];
  }
  asm volatile("s_wait_asynccnt 0" ::: "memory");
  // Each lane reads back exactly the LDS chunks it issued -> wave-local dep,
  // no workgroup barrier needed.
#pragma unroll
  for (int i = 0; i < 4; ++i) {
    const int slot = tid + i * 256;
    const v4f r = smem[slot];
    const v4f o = r + p[i] + e[i];
    __builtin_nontemporal_store(o, (v4f*)(outRow + slot * 4));
  }
#else
  // Portable fallback (host parse / non-gfx1250): direct vector loads
  const v4f* rowV = (const v4f*)rowBase;
#pragma unroll
  for (int i = 0; i < 4; ++i) {
    const int slot = tid + i * 256;
    const v4f o = rowV[slot] + posRow[slot] + edgeRow[slot];
    __builtin_nontemporal_store(o, (v4f*)(outRow + slot * 4));
  }
  (void)smem;
#endif
}

// ---------------------------------------------------------------------------
extern "C" void kernel_launch(void* const* d_in, const int* in_sizes, int n_in,
                              void* d_out, int out_size, void* d_ws, size_t ws_size,
                              hipStream_t stream) {
  const int*   x    = (const int*)d_in[0];    // [B, SEQ] int32
  const float* W    = (const float*)d_in[1];  // [8, 6300, 4096] f32
  const float* bias = (const float*)d_in[2];  // [4096] f32
  const float* pos  = (const float*)d_in[3];  // [2048, 4096] f32
  float* out  = (float*)d_out;                // [B, SEQ, 4096] f32
  float* edge = (float*)d_ws;                 // [8, 4096] f32 scratch (128 KB)

  edge_precompute<<<DMODEL / 128, 128, 0, stream>>>(W, bias, edge);
  gather_add<<<NBATCH * SEQLEN, 256, 0, stream>>>(x, W, pos, edge, out);
}